// SAGEBackbone_10720238371194
// MI455X (gfx1250) — compile-verified
//
#include <hip/hip_runtime.h>

#define D 128
#define DD (D * D)

typedef float v2f __attribute__((ext_vector_type(2)));
typedef float v8f __attribute__((ext_vector_type(8)));

// ---------------------------------------------------------------------------
// Native fp32 global atomic add (no return) at device scope.
// msg_sum (51.2 MB) fits in the 192 MB L2, so these resolve in L2 atomic units.
// ---------------------------------------------------------------------------
__device__ __forceinline__ void gatomic_add_f32(float* p, float v) {
    asm volatile("global_atomic_add_f32 %0, %1, off scope:SCOPE_DEV"
                 :: "v"(p), "v"(v) : "memory");
}

// ---------------------------------------------------------------------------
// Zero a contiguous float range (grid-stride).
// ---------------------------------------------------------------------------
__global__ __launch_bounds__(256)
void zero_kernel(float* __restrict__ p, long n) {
    long stride = (long)gridDim.x * blockDim.x;
    for (long i = (long)blockIdx.x * blockDim.x + threadIdx.x; i < n; i += stride)
        p[i] = 0.0f;
}

// ---------------------------------------------------------------------------
// Transpose one 128x128 weight: Wt[k*128+n] = W[n*128+k]  (B = W^T row-major)
// ---------------------------------------------------------------------------
__global__ __launch_bounds__(256)
void transpose128_kernel(const float* __restrict__ W, float* __restrict__ Wt) {
    int idx = blockIdx.x * 256 + threadIdx.x;   // 64 blocks cover 16384 elems
    if (idx < DD) {
        int r = idx >> 7, c = idx & 127;
        Wt[c * D + r] = W[idx];
    }
}

// ---------------------------------------------------------------------------
// Edge scatter: one wave per edge. Each lane gathers float4 of h[src] and does
// 4 native f32 atomic adds into msg[dst]; lane 0 bumps the degree counter.
// ---------------------------------------------------------------------------
__global__ __launch_bounds__(256)
void sage_scatter_kernel(const float* __restrict__ h,
                         const int* __restrict__ src,
                         const int* __restrict__ dst,
                         float* __restrict__ msg,
                         float* __restrict__ cnt,
                         int n_edges) {
    int wave = threadIdx.x >> 5;
    int lane = threadIdx.x & 31;
    int e = blockIdx.x * 8 + wave;
    if (e >= n_edges) return;

    int s = src[e];
    int d = dst[e];

    const float4 v = *(const float4*)(h + (long)s * D + lane * 4);
    float* p = msg + (long)d * D + lane * 4;
    gatomic_add_f32(p + 0, v.x);
    gatomic_add_f32(p + 1, v.y);
    gatomic_add_f32(p + 2, v.z);
    gatomic_add_f32(p + 3, v.w);
    if (lane == 0) gatomic_add_f32(cnt + d, 1.0f);
}

// ---------------------------------------------------------------------------
// Fused SAGE linear:  out = (msg / max(cnt,1)) @ Wl^T + b + h @ Wr^T  [+ReLU]
// 8 waves/block, 16 rows/wave. K-outer loop: each A fragment (msg row / h row)
// is loaded from global ONCE and reused across all 8 output N-tiles, whose
// accumulators (8 x v8f = 64 VGPRs) live in registers. Back-to-back WMMAs hit
// different accumulators -> no D->C RAW chain stalls in the XDL pipe.
// Weights (already transposed) staged in 128 KB of LDS.
// ---------------------------------------------------------------------------
__global__ __launch_bounds__(256)
void sage_gemm_kernel(const float* __restrict__ msg,
                      const float* __restrict__ cnt,
                      const float* __restrict__ h,
                      const float* __restrict__ Wtl,   // [k][n] = Wl[n][k]
                      const float* __restrict__ Wtr,   // [k][n] = Wr[n][k]
                      const float* __restrict__ bias,
                      float* __restrict__ out,
                      int n_nodes, int do_relu) {
    __shared__ float ldsWl[DD];
    __shared__ float ldsWr[DD];

    int tid = threadIdx.x;
    {   // cooperative stage of both transposed weights into LDS (float4)
        const float4* gl = (const float4*)Wtl;
        const float4* gr = (const float4*)Wtr;
        float4* ll = (float4*)ldsWl;
        float4* lr = (float4*)ldsWr;
        for (int i = tid; i < DD / 4; i += 256) { ll[i] = gl[i]; lr[i] = gr[i]; }
    }
    __syncthreads();

    int wave = tid >> 5;
    int lane = tid & 31;
    int m    = lane & 15;                  // row within 16-row tile
    int koff = (lane < 16) ? 0 : 2;        // A-matrix 16x4 layout: k-half by lane group

    long row0 = (long)blockIdx.x * 128 + wave * 16;
    if (row0 >= n_nodes) return;           // uniform per wave: EXEC stays all-1s
    long row = row0 + m;

    float invc = 1.0f / fmaxf(cnt[row], 1.0f);
    const float* msgRow = msg + row * D;
    const float* hRow   = h   + row * D;

    v8f acc[8];
    #pragma unroll
    for (int nt = 0; nt < 8; ++nt)
        acc[nt] = (v8f){0.f, 0.f, 0.f, 0.f, 0.f, 0.f, 0.f, 0.f};

    #pragma unroll 2
    for (int ks = 0; ks < 32; ++ks) {
        int k0 = ks * 4 + koff;
        float2 am = *(const float2*)(msgRow + k0);   // A loaded once per k-step
        float2 ah = *(const float2*)(hRow + k0);
        v2f aAgg  = { am.x * invc, am.y * invc };
        v2f aSelf = { ah.x, ah.y };
        const float* wl0 = ldsWl + k0 * D + m;
        const float* wl1 = ldsWl + (k0 + 1) * D + m;
        const float* wr0 = ldsWr + k0 * D + m;
        const float* wr1 = ldsWr + (k0 + 1) * D + m;

        #pragma unroll
        for (int nt = 0; nt < 8; ++nt) {
            int nn = nt * 16;
            v2f bl = { wl0[nn], wl1[nn] };
            v2f br = { wr0[nn], wr1[nn] };
            acc[nt] = __builtin_amdgcn_wmma_f32_16x16x4_f32(
                false, aAgg, false, bl, (short)0, acc[nt], false, false);
            acc[nt] = __builtin_amdgcn_wmma_f32_16x16x4_f32(
                false, aSelf, false, br, (short)0, acc[nt], false, false);
        }
    }

    int mbase = (lane >= 16) ? 8 : 0;      // C/D layout: vgpr v -> M = v (+8 hi lanes)
    for (int nt = 0; nt < 8; ++nt) {
        float bn = bias[nt * 16 + m];
        #pragma unroll
        for (int v = 0; v < 8; ++v) {
            float val = acc[nt][v] + bn;
            if (do_relu) val = fmaxf(val, 0.0f);
            out[(row0 + v + mbase) * D + nt * 16 + m] = val;
        }
    }
}

// ---------------------------------------------------------------------------
// Row-wise L2 normalize: one wave per node row (wave32), float4 per lane.
// ---------------------------------------------------------------------------
__global__ __launch_bounds__(256)
void l2norm_kernel(float* __restrict__ out, int n_nodes) {
    int wave = threadIdx.x >> 5;
    int lane = threadIdx.x & 31;
    long row = (long)blockIdx.x * 8 + wave;
    if (row >= n_nodes) return;

    float4* p = (float4*)(out + row * D);
    float4 a = p[lane];
    float ss = a.x * a.x + a.y * a.y + a.z * a.z + a.w * a.w;
    #pragma unroll
    for (int off = 16; off >= 1; off >>= 1)
        ss += __shfl_xor(ss, off, 32);
    float scale = 1.0f / fmaxf(sqrtf(ss), 1e-12f);
    a.x *= scale; a.y *= scale; a.z *= scale; a.w *= scale;
    p[lane] = a;
}

// ---------------------------------------------------------------------------
extern "C" void kernel_launch(void* const* d_in, const int* in_sizes, int n_in,
                              void* d_out, int out_size, void* d_ws, size_t ws_size,
                              hipStream_t stream) {
    const float* x   = (const float*)d_in[0];
    const int*   ei  = (const int*)d_in[1];
    const float* W1l = (const float*)d_in[2];
    const float* b1  = (const float*)d_in[3];
    const float* W1r = (const float*)d_in[4];
    const float* W2l = (const float*)d_in[5];
    const float* b2  = (const float*)d_in[6];
    const float* W2r = (const float*)d_in[7];

    const int N = in_sizes[0] / D;       // 100000
    const int E = in_sizes[1] / 2;       // 1600000
    const int* src = ei;
    const int* dst = ei + E;

    // workspace layout (floats): msg[N*D] | cnt[N] | h1[N*D] | 4x Wt[128*128]
    float* ws   = (float*)d_ws;
    float* msg  = ws;
    float* cnt  = msg + (size_t)N * D;
    float* h1   = cnt + N;
    float* Wt1l = h1 + (size_t)N * D;
    float* Wt1r = Wt1l + DD;
    float* Wt2l = Wt1r + DD;
    float* Wt2r = Wt2l + DD;
    float* out  = (float*)d_out;

    // pre-transpose weights so B = W^T is row-major for LDS staging
    transpose128_kernel<<<64, 256, 0, stream>>>(W1l, Wt1l);
    transpose128_kernel<<<64, 256, 0, stream>>>(W1r, Wt1r);
    transpose128_kernel<<<64, 256, 0, stream>>>(W2l, Wt2l);
    transpose128_kernel<<<64, 256, 0, stream>>>(W2r, Wt2r);

    const long zcount = (long)N * D + N;
    const int gemmBlocks = (N + 127) / 128;
    const int scatBlocks = (E + 7) / 8;

    // ---- layer 1 ----
    zero_kernel<<<2048, 256, 0, stream>>>(msg, zcount);
    sage_scatter_kernel<<<scatBlocks, 256, 0, stream>>>(x, src, dst, msg, cnt, E);
    sage_gemm_kernel<<<gemmBlocks, 256, 0, stream>>>(msg, cnt, x, Wt1l, Wt1r, b1,
                                                     h1, N, /*relu=*/1);
    // ---- layer 2 ----
    zero_kernel<<<2048, 256, 0, stream>>>(msg, zcount);
    sage_scatter_kernel<<<scatBlocks, 256, 0, stream>>>(h1, src, dst, msg, cnt, E);
    sage_gemm_kernel<<<gemmBlocks, 256, 0, stream>>>(msg, cnt, h1, Wt2l, Wt2r, b2,
                                                     out, N, /*relu=*/0);
    // ---- final L2 normalize ----
    l2norm_kernel<<<(N + 7) / 8, 256, 0, stream>>>(out, N);
}